// FastRCNN_12610023981676
// MI455X (gfx1250) — compile-verified
//
#include <hip/hip_runtime.h>
#include <hip/hip_bf16.h>

// ---------------- problem constants (from reference) ----------------
#define NIMG   8
#define PPN    4000
#define NCLS   91
#define DET_K  100
#define IMG_W  800.0f
#define IMG_H  800.0f
#define NEGV   (-1000000000.0f)
#define NEG_HALF (-500000000.0f)
#define CLAMPV 4.135166556742356f   // log(1000/16)

typedef __attribute__((ext_vector_type(4))) unsigned int v4u;
typedef __attribute__((ext_vector_type(8))) int          v8i;
typedef __attribute__((ext_vector_type(4))) int          v4i;

// -------------------------------------------------------------------
// TDM: DMA `n` dwords (n <= 65535) from global memory into LDS.
// Descriptor per CDNA5 ISA ch.8 (CDNA5 ISA): 2D tensor,
// tile_dim0 = n, tile_dim1 = 1, data_size = 4B, no pad/iterate/multicast.
// Issue from one full wave; completion via s_wait_tensorcnt.
// 6-arg builtin (clang-23 / therock-10.0); trailing int32x8 zero-filled.
// -------------------------------------------------------------------
__device__ __forceinline__ void tdm_load_dwords(const void* gsrc, void* ldst,
                                                unsigned n) {
  unsigned long long ga = (unsigned long long)(uintptr_t)gsrc;
  unsigned lds = (unsigned)(uintptr_t)ldst;  // low 32 bits of flat ptr = LDS byte offset

  v4u g0;
  g0.x = 1u;                                   // count=1, is_restore=0, gather off
  g0.y = lds;                                  // lds_addr
  g0.z = (unsigned)(ga & 0xffffffffu);         // global_addr[31:0]
  g0.w = (unsigned)((ga >> 32) & 0x01ffffffu)  // global_addr[56:32]
       | (2u << 30);                           // type = 2 ("image")

  v8i g1;
  g1[0] = (int)(2u << 16);                     // wg_mask=0, data_size=2 (4 bytes)
  g1[1] = (int)((n & 0xffffu) << 16);          // tensor_dim0[15:0]
  g1[2] = (int)(((n >> 16) & 0xffffu)          // tensor_dim0[31:16]
       | (1u << 16));                          // tensor_dim1 = 1
  g1[3] = (int)((n & 0xffffu) << 16);          // tensor_dim1 hi = 0, tile_dim0 = n
  g1[4] = 1;                                   // tile_dim1 = 1, tile_dim2 = 0
  g1[5] = (int)n;                              // tensor_dim0_stride[31:0]
  g1[6] = 0;                                   // stride0 hi, stride1 lo
  g1[7] = 0;                                   // stride1 hi

  v4i g2;                                      // normal mode, dims 2/3 unused
  g2[0] = 1;                                   // tensor_dim2
  g2[1] = 1;                                   // tensor_dim3 (iterate_enable=0)
  g2[2] = 0;                                   // tensor_dim2_stride lo
  g2[3] = 0;                                   // stride hi, tile_dim3 = 0

  v4i g3;
  g3[0] = 0;                                   // tensor_dim3_stride lo
  g3[1] = (int)(1u << 16);                     // stride hi=0, tensor_dim4[15:0]=1
  g3[2] = 0;                                   // tensor_dim4 hi, tile_dim4 = 0
  g3[3] = 0;

  v8i gx = {0, 0, 0, 0, 0, 0, 0, 0};           // extra group (zero-filled)

  __builtin_amdgcn_tensor_load_to_lds(g0, g1, g2, g3, gx, /*cpol=*/0);
}

// -------------------------------------------------------------------
// Phase 1: per-proposal softmax / argmax / box decode.
// One thread per proposal. Branchless one-pass online softmax:
//   e = exp(-|x - m|);  m' = max(m,x);  s' = (x>m) ? s*e+1 : s+e
// -> one v_exp + cndmasks per class, no exec-mask divergence.
// -------------------------------------------------------------------
__global__ void score_decode_kernel(const float* __restrict__ logits,
                                    const float* __restrict__ deltas,
                                    const float* __restrict__ props,
                                    float* __restrict__ g_boxes,
                                    float* __restrict__ g_act,
                                    int*   __restrict__ g_labels) {
  int i = blockIdx.x * blockDim.x + threadIdx.x;
  if (i >= NIMG * PPN) return;

  const float* lp = logits + (size_t)i * NCLS;

  float m = lp[0];           // running max (class 0 included in denominator)
  float s = 1.0f;            // running sum of exp(x - m)
  float fgmax = -3.0e38f;    // max foreground logit
  int   lab = 1;
  for (int c = 1; c < NCLS; ++c) {
    float x = lp[c];
    float e = __expf(-fabsf(x - m));
    bool  gt = x > m;
    s = gt ? __fmaf_rn(s, e, 1.0f) : (s + e);
    m = fmaxf(m, x);
    // strict > keeps first occurrence == jnp.argmax tie-break
    lab   = (x > fgmax) ? c : lab;
    fgmax = fmaxf(fgmax, x);
  }
  float score = __expf(fgmax - m) / s;

  // gather the winning class's 4 deltas
  const float* dp = deltas + (size_t)i * (NCLS * 4) + lab * 4;
  float dx = dp[0] * 0.1f;                       // / wx(10)
  float dy = dp[1] * 0.1f;                       // / wy(10)
  float dw = fminf(dp[2] * 0.2f, CLAMPV);        // / ww(5), clamp
  float dh = fminf(dp[3] * 0.2f, CLAMPV);        // / wh(5), clamp

  const float* pp = props + (size_t)i * 4;
  float x1 = pp[0], y1 = pp[1], x2 = pp[2], y2 = pp[3];
  float w  = x2 - x1, h = y2 - y1;
  float cx = x1 + 0.5f * w, cy = y1 + 0.5f * h;
  float pcx = dx * w + cx, pcy = dy * h + cy;
  float pw  = __expf(dw) * w, ph = __expf(dh) * h;

  float bx1 = fminf(fmaxf(pcx - 0.5f * pw, 0.0f), IMG_W);
  float by1 = fminf(fmaxf(pcy - 0.5f * ph, 0.0f), IMG_H);
  float bx2 = fminf(fmaxf(pcx + 0.5f * pw, 0.0f), IMG_W);
  float by2 = fminf(fmaxf(pcy + 0.5f * ph, 0.0f), IMG_H);

  float ws = bx2 - bx1, hs = by2 - by1;
  bool valid = (ws >= 0.01f) && (hs >= 0.01f) && (score > 0.05f);

  float* bo = g_boxes + (size_t)i * 4;
  bo[0] = bx1; bo[1] = by1; bo[2] = bx2; bo[3] = by2;
  g_act[i]    = valid ? score : NEGV;
  g_labels[i] = lab;
}

// -------------------------------------------------------------------
// Phase 2: per-image greedy NMS, 100 rounds, entirely out of LDS.
// One 1024-thread (32-wave) block per image. TDM stages boxes+act
// into LDS once; argmax uses wave32 shuffles + LDS cross-wave reduce.
// -------------------------------------------------------------------
__global__ void __launch_bounds__(1024)
nms_kernel(const float* __restrict__ g_boxes,
           const float* __restrict__ g_act,
           const int*   __restrict__ g_labels,
           float* __restrict__ out) {
  extern __shared__ float smem[];
  float* s_ob  = smem;               // 4*PPN floats (boxes, later offset boxes)
  float* s_act = smem + 4 * PPN;     // PPN floats

  __shared__ float s_rv[32];
  __shared__ int   s_ri[32];
  __shared__ float s_bestv;
  __shared__ int   s_besti;
  __shared__ float s_maxc;
  __shared__ int   s_idx[DET_K];
  __shared__ float s_val[DET_K];
  __shared__ int   s_keep[DET_K];

  const int n   = blockIdx.x;
  const int tid = threadIdx.x;
  const int nth = blockDim.x;

  const float* bsrc = g_boxes  + (size_t)n * PPN * 4;
  const float* asrc = g_act    + (size_t)n * PPN;
  const int*   lsrc = g_labels + (size_t)n * PPN;

  // --- TDM stage: boxes (64 KB) + act (16 KB) -> LDS, wave 0 only ---
  if (tid < 32) {
    tdm_load_dwords(bsrc, s_ob, PPN * 4);
    tdm_load_dwords(asrc, s_act, PPN);
    __builtin_amdgcn_s_wait_tensorcnt(0);
  }
  __syncthreads();

  // --- maxc = max over valid boxes of all coords (0 if none) ---
  float lm = 0.0f;
  for (int j = tid; j < PPN; j += nth) {
    if (s_act[j] > NEG_HALF) {
      float a = fmaxf(s_ob[4 * j + 0], s_ob[4 * j + 1]);
      float b = fmaxf(s_ob[4 * j + 2], s_ob[4 * j + 3]);
      lm = fmaxf(lm, fmaxf(a, b));
    }
  }
  for (int off = 16; off > 0; off >>= 1) lm = fmaxf(lm, __shfl_down(lm, off, 32));
  if ((tid & 31) == 0) s_rv[tid >> 5] = lm;
  __syncthreads();
  if (tid < 32) {
    float v = s_rv[tid];
    for (int off = 16; off > 0; off >>= 1) v = fmaxf(v, __shfl_down(v, off, 32));
    if (tid == 0) s_maxc = v;
  }
  __syncthreads();

  // --- per-class offset: ob = box + label * (maxc + 1) ---
  const float offm = s_maxc + 1.0f;
  for (int j = tid; j < PPN; j += nth) {
    float o = (float)lsrc[j] * offm;
    s_ob[4 * j + 0] += o;
    s_ob[4 * j + 1] += o;
    s_ob[4 * j + 2] += o;
    s_ob[4 * j + 3] += o;
  }
  __syncthreads();

  // --- greedy NMS: 100 rounds of argmax + suppress ---
  for (int t = 0; t < DET_K; ++t) {
    // block argmax (max value, smallest index on ties == jnp.argmax)
    float bv = -3.0e38f;
    int   bi = 0x7fffffff;
    for (int j = tid; j < PPN; j += nth) {
      float v = s_act[j];
      if (v > bv || (v == bv && j < bi)) { bv = v; bi = j; }
    }
    for (int off = 16; off > 0; off >>= 1) {
      float ov = __shfl_down(bv, off, 32);
      int   oi = __shfl_down(bi, off, 32);
      if (ov > bv || (ov == bv && oi < bi)) { bv = ov; bi = oi; }
    }
    if ((tid & 31) == 0) { s_rv[tid >> 5] = bv; s_ri[tid >> 5] = bi; }
    __syncthreads();
    if (tid < 32) {
      float v  = s_rv[tid];
      int   ii = s_ri[tid];
      for (int off = 16; off > 0; off >>= 1) {
        float ov = __shfl_down(v, off, 32);
        int   oi = __shfl_down(ii, off, 32);
        if (ov > v || (ov == v && oi < ii)) { v = ov; ii = oi; }
      }
      if (tid == 0) { s_bestv = v; s_besti = ii; }
    }
    __syncthreads();

    const int   bi2  = s_besti;
    const float bv2  = s_bestv;
    const int   keep = (bv2 > NEG_HALF) ? 1 : 0;

    if (tid == 0) {
      s_idx[t]  = bi2;
      s_val[t]  = bv2;
      s_keep[t] = keep;
      s_act[bi2] = NEGV;   // winner always removed (suppression writes same value)
    }
    if (keep) {
      const float qx1 = s_ob[4 * bi2 + 0];
      const float qy1 = s_ob[4 * bi2 + 1];
      const float qx2 = s_ob[4 * bi2 + 2];
      const float qy2 = s_ob[4 * bi2 + 3];
      const float qa  = (qx2 - qx1) * (qy2 - qy1);
      for (int j = tid; j < PPN; j += nth) {
        float cx1 = s_ob[4 * j + 0];
        float cy1 = s_ob[4 * j + 1];
        float cx2 = s_ob[4 * j + 2];
        float cy2 = s_ob[4 * j + 3];
        float ix1 = fmaxf(qx1, cx1), iy1 = fmaxf(qy1, cy1);
        float ix2 = fminf(qx2, cx2), iy2 = fminf(qy2, cy2);
        float inter = fmaxf(ix2 - ix1, 0.0f) * fmaxf(iy2 - iy1, 0.0f);
        float a2 = (cx2 - cx1) * (cy2 - cy1);
        float iou = inter / fmaxf(qa + a2 - inter, 1e-9f);
        if (iou > 0.5f) s_act[j] = NEGV;
      }
    }
    __syncthreads();
  }

  // --- outputs: flat tuple concat (labels[8,100], scores[8,100], boxes[8,100,4]) ---
  if (tid < DET_K) {
    int i = s_idx[tid];
    int k = s_keep[tid];
    int o = n * DET_K + tid;
    out[o]                    = k ? (float)lsrc[i] : 0.0f;        // labels
    out[NIMG * DET_K + o]     = k ? s_val[tid]     : 0.0f;        // scores
    float* bo = out + 2 * NIMG * DET_K + (size_t)o * 4;           // boxes
    bo[0] = k ? bsrc[4 * i + 0] : 0.0f;
    bo[1] = k ? bsrc[4 * i + 1] : 0.0f;
    bo[2] = k ? bsrc[4 * i + 2] : 0.0f;
    bo[3] = k ? bsrc[4 * i + 3] : 0.0f;
  }
}

// -------------------------------------------------------------------
extern "C" void kernel_launch(void* const* d_in, const int* in_sizes, int n_in,
                              void* d_out, int out_size, void* d_ws, size_t ws_size,
                              hipStream_t stream) {
  // inputs (setup_inputs order): images, class_logits, pred_bbox_deltas, proposals
  const float* logits = (const float*)d_in[1];
  const float* deltas = (const float*)d_in[2];
  const float* props  = (const float*)d_in[3];

  // workspace layout: boxes | act | labels  (~768 KB total)
  float* g_boxes = (float*)d_ws;                         // NIMG*PPN*4 f32
  float* g_act   = g_boxes + (size_t)NIMG * PPN * 4;     // NIMG*PPN   f32
  int*   g_lab   = (int*)(g_act + (size_t)NIMG * PPN);   // NIMG*PPN   i32

  const int total = NIMG * PPN;
  score_decode_kernel<<<(total + 255) / 256, 256, 0, stream>>>(
      logits, deltas, props, g_boxes, g_act, g_lab);

  const size_t lds_bytes = (size_t)(4 * PPN + PPN) * sizeof(float);  // 80 000 B
  nms_kernel<<<NIMG, 1024, lds_bytes, stream>>>(g_boxes, g_act, g_lab,
                                                (float*)d_out);
}